// MambaMoEBlock_67577015435317
// MI455X (gfx1250) — compile-verified
//
#include <hip/hip_runtime.h>
#include <hip/hip_bf16.h>

// ---------------------------------------------------------------------------
// MambaMoE block for gfx1250 (MI455X). All fp32. GEMMs use V_WMMA_F32_16X16X4_F32.
// ---------------------------------------------------------------------------

#define NB   4
#define C1   96
#define C2   96
#define HH   40
#define WW   40
#define LL   1600          // H*W
#define DE   192
#define DI   192
#define RR   6
#define NN   16
#define KK   4
#define EE   4

typedef float v2f __attribute__((ext_vector_type(2)));
typedef float v8f __attribute__((ext_vector_type(8)));

enum {
  EPI_NONE = 0,
  EPI_BNSILU,        // val = val*g[m]*rsqrt(1+eps) + b[m]; silu
  EPI_BIAS_GELU,     // val = gelu(val + p0[m])
  EPI_BIAS_RES,      // val = val + p0[m] + res
  EPI_RES,           // val = val + res
  EPI_BIAS_SOFTPLUS, // val = softplus(val + p0[m])
  EPI_MOE            // Y += wmat[b,e] * (val + p0[m])
};

__device__ __forceinline__ float gelu_f(float x) {
  return 0.5f * x * (1.f + erff(x * 0.70710678118654752f));
}
__device__ __forceinline__ float silu_f(float x) {
  return x / (1.f + __expf(-x));
}
__device__ __forceinline__ float softplus_f(float x) {
  return (x > 20.f) ? x : log1pf(__expf(x));
}

// ---------------------------------------------------------------------------
// fp32 WMMA GEMM: Y[b, m, p] = sum_k Wm[m,k] * X[b, k, p]   (+ epilogue)
// Row stride inside a batch is compile-time LL=1600 -> all K-loop loads are
// base + immediate-offset. One wave computes a 16x32 tile (two 16x16 WMMA
// accumulators sharing the A fragment).  Column tiles (32 px) never cross a
// batch boundary since 1600 % 32 == 0.
//
// Fragments per ISA 7.12.2 (wave32):
//   A 16x4 : lane -> M = lane%16 ; VGPR i, half = lane>>4 -> K = 2*half + i
//   B 4x16 : VGPR v -> K = 2*half + v, N = lane%16 (symmetric half-split)
//   C/D    : VGPR v -> M = v + 8*half, N = lane%16
//
// MG=true: A row index clamped (branchless); stores guarded. K tail (KD%4!=0)
// handled with clamped loads + select-to-zero (branchless).
// ---------------------------------------------------------------------------
template <int KD, bool MG>
__global__ __launch_bounds__(256) void mmb_wmma_gemm(
    const float* __restrict__ Wm, const float* __restrict__ X,
    float* __restrict__ Y, int M, long sXb, long sYb, int epi,
    const float* __restrict__ p0, const float* __restrict__ p1,
    const float* __restrict__ res, long sResB,
    const float* __restrict__ wsc, int eidx) {
  constexpr int LP = LL;
  const int lane = threadIdx.x & 31;
  const int wave = threadIdx.x >> 5;
  const int Mt = (M + 15) >> 4;
  const int Nt = (NB * LP) >> 5;  // 32-wide column tiles
  const int tile = blockIdx.x * (blockDim.x >> 5) + wave;
  if (tile >= Mt * Nt) return;  // uniform per wave -> EXEC stays full
  const int tm = tile / Nt;
  const int tn = tile % Nt;
  const int P0 = tn << 5;
  const int b = P0 / LP;
  const int pix = P0 % LP;
  const int row = lane & 15;
  const int half = lane >> 4;

  int ma = tm * 16 + row;
  if (MG) ma = (ma < M) ? ma : (M - 1);  // clamp: rows >= M computed, never stored
  const float* Ap = Wm + (long)ma * KD + 2 * half;
  const float* Bp = X + (long)b * sXb + pix + (long)(2 * half) * LP + row;

  v8f acc0 = {};
  v8f acc1 = {};
  constexpr int K4 = KD & ~3;
#pragma unroll
  for (int k0 = 0; k0 < K4; k0 += 4) {
    v2f av = *(const v2f*)(Ap + k0);
    v2f b0v, b1v;
    b0v.x = Bp[(long)k0 * LP];
    b0v.y = Bp[(long)(k0 + 1) * LP];
    b1v.x = Bp[(long)k0 * LP + 16];
    b1v.y = Bp[(long)(k0 + 1) * LP + 16];
    acc0 = __builtin_amdgcn_wmma_f32_16x16x4_f32(false, av, false, b0v,
                                                 (short)0, acc0, false, false);
    acc1 = __builtin_amdgcn_wmma_f32_16x16x4_f32(false, av, false, b1v,
                                                 (short)0, acc1, false, false);
  }
  if constexpr (K4 < KD) {
    // tail K-step: lanes with k >= KD contribute zero (A selected to 0).
    const int ka = K4 + 2 * half;
    const int kc = (ka <= KD - 2) ? ka : (KD - 2);  // clamped, in-bounds
    v2f av = *(const v2f*)(Wm + (long)ma * KD + kc);
    av.x = (ka < KD) ? av.x : 0.f;
    av.y = (ka + 1 < KD) ? av.y : 0.f;
    const float* Bq = X + (long)b * sXb + pix + (long)kc * LP + row;
    v2f b0v, b1v;
    b0v.x = Bq[0];
    b0v.y = Bq[LP];
    b1v.x = Bq[16];
    b1v.y = Bq[LP + 16];
    acc0 = __builtin_amdgcn_wmma_f32_16x16x4_f32(false, av, false, b0v,
                                                 (short)0, acc0, false, false);
    acc1 = __builtin_amdgcn_wmma_f32_16x16x4_f32(false, av, false, b1v,
                                                 (short)0, acc1, false, false);
  }

  float* Yb = Y + (long)b * sYb + pix + row;
  const float* Rb = res + (long)b * sResB + pix + row;  // only deref'd if used
#pragma unroll
  for (int v = 0; v < 8; ++v) {
    const int m = tm * 16 + v + 8 * half;
    if (MG && m >= M) continue;
    const long off = (long)m * LP;
    float v0 = acc0[v];
    float v1 = acc1[v];
    switch (epi) {
      case EPI_BNSILU: {
        float s = p0[m] * rsqrtf(1.f + 1e-5f);
        v0 = silu_f(v0 * s + p1[m]);
        v1 = silu_f(v1 * s + p1[m]);
      } break;
      case EPI_BIAS_GELU: {
        float bi = p0[m];
        v0 = gelu_f(v0 + bi);
        v1 = gelu_f(v1 + bi);
      } break;
      case EPI_BIAS_RES: {
        float bi = p0[m];
        v0 += bi + Rb[off];
        v1 += bi + Rb[off + 16];
      } break;
      case EPI_RES:
        v0 += Rb[off];
        v1 += Rb[off + 16];
        break;
      case EPI_BIAS_SOFTPLUS: {
        float bi = p0[m];
        v0 = softplus_f(v0 + bi);
        v1 = softplus_f(v1 + bi);
      } break;
      case EPI_MOE: {
        float wgt = wsc[b * EE + eidx];
        float bi = p0[m];
        v0 = Yb[off] + wgt * (v0 + bi);
        v1 = Yb[off + 16] + wgt * (v1 + bi);
      } break;
      default:
        break;
    }
    Yb[off] = v0;
    Yb[off + 16] = v1;
  }
}

// ---------------------------------------------------------------------------
// Depthwise 3x3 SAME conv + bias, optional BN (g/sqrt(1+eps), b) and GELU.
// ---------------------------------------------------------------------------
__global__ void mmb_dwconv3(const float* __restrict__ in, float* __restrict__ out,
                            const float* __restrict__ w, const float* __restrict__ bias,
                            const float* __restrict__ bng, const float* __restrict__ bnb,
                            int C, long inB, long outB, int dogelu, int total) {
  int idx = blockIdx.x * blockDim.x + threadIdx.x;
  if (idx >= total) return;
  int l = idx % LL;
  int c = (idx / LL) % C;
  int b = idx / (LL * C);
  int hy = l / WW, wx = l % WW;
  const float* ip = in + (long)b * inB + (long)c * LL;
  const float* wp = w + c * 9;
  float s = 0.f;
#pragma unroll
  for (int dy = 0; dy < 3; ++dy) {
    int yy = hy + dy - 1;
    if (yy < 0 || yy >= HH) continue;
#pragma unroll
    for (int dx = 0; dx < 3; ++dx) {
      int xx = wx + dx - 1;
      if (xx < 0 || xx >= WW) continue;
      s += wp[dy * 3 + dx] * ip[yy * WW + xx];
    }
  }
  s += bias[c];
  if (bng) s = s * (bng[c] * rsqrtf(1.f + 1e-5f)) + bnb[c];
  if (dogelu) s = gelu_f(s);
  out[(long)b * outB + (long)c * LL + l] = s;
}

// BN eval: y = x * g/sqrt(1+eps) + b   ((B,C,L) layout)
__global__ void mmb_bn(const float* __restrict__ in, float* __restrict__ out,
                       const float* __restrict__ g, const float* __restrict__ b,
                       int C, int total) {
  int idx = blockIdx.x * blockDim.x + threadIdx.x;
  if (idx >= total) return;
  int c = (idx / LL) % C;
  out[idx] = in[idx] * (g[c] * rsqrtf(1.f + 1e-5f)) + b[c];
}

// z1 = gelu(xz[:, DE:2DE, :])
__global__ void mmb_geluz(const float* __restrict__ xz, float* __restrict__ z1, int total) {
  int idx = blockIdx.x * blockDim.x + threadIdx.x;
  if (idx >= total) return;
  int l = idx % LL;
  int d = (idx / LL) % DE;
  int b = idx / (LL * DE);
  z1[idx] = gelu_f(xz[((long)b * 2 * DE + DE + d) * LL + l]);
}

// xs[b,k,d,l] for the 4 scan directions from xx (B,DE,L)
__global__ void mmb_build_xs(const float* __restrict__ xx, float* __restrict__ xs, int total) {
  int idx = blockIdx.x * blockDim.x + threadIdx.x;
  if (idx >= total) return;
  int l = idx % LL;
  int d = (idx / LL) % DE;
  int b = idx / (LL * DE);
  int hy = l / WW, wx = l % WW;
  int lt = wx * HH + hy;  // transposed index
  float v = xx[((long)b * DE + d) * LL + l];
  long base = (long)b * KK * DE * LL + (long)d * LL;
  xs[base + 0L * DE * LL + l] = v;
  xs[base + 1L * DE * LL + lt] = v;
  xs[base + 2L * DE * LL + (LL - 1 - l)] = v;
  xs[base + 3L * DE * LL + (LL - 1 - lt)] = v;
}

// As = -exp(A_logs)   (K*DI*N flat)
__global__ void mmb_make_as(const float* __restrict__ al, float* __restrict__ As, int total) {
  int idx = blockIdx.x * blockDim.x + threadIdx.x;
  if (idx >= total) return;
  As[idx] = -__expf(al[idx]);
}

// Selective scan: grid = B*K blocks of DI threads, N=16 states in registers.
// ys[b,k,d,l] = sum_n h*C + Ds[k,d]*x
__global__ void mmb_scan(const float* __restrict__ xs, const float* __restrict__ dts,
                         const float* __restrict__ xdbl, const float* __restrict__ As,
                         const float* __restrict__ Ds, float* __restrict__ ys) {
  int bk = blockIdx.x;
  int b = bk >> 2, k = bk & 3;
  int d = threadIdx.x;
  const float* xp = xs + ((long)(b * KK + k) * DI + d) * LL;
  const float* dtp = dts + ((long)(b * KK + k) * DI + d) * LL;
  const float* Bp = xdbl + ((long)b * KK + k) * (RR + 2 * NN) * LL + (long)RR * LL;
  const float* Cp = Bp + (long)NN * LL;
  float* yp = ys + ((long)(b * KK + k) * DI + d) * LL;
  float A[NN];
#pragma unroll
  for (int n = 0; n < NN; ++n) A[n] = As[(k * DI + d) * NN + n];
  float Dv = Ds[k * DI + d];
  float h[NN];
#pragma unroll
  for (int n = 0; n < NN; ++n) h[n] = 0.f;
  __shared__ float sB[NN], sC[NN];
  for (int l = 0; l < LL; ++l) {
    if (threadIdx.x < NN) sB[threadIdx.x] = Bp[(long)threadIdx.x * LL + l];
    else if (threadIdx.x < 2 * NN) sC[threadIdx.x - NN] = Cp[(long)(threadIdx.x - NN) * LL + l];
    __syncthreads();
    float dt = dtp[l];
    float x = xp[l];
    float dtx = dt * x;
    float y = 0.f;
#pragma unroll
    for (int n = 0; n < NN; ++n) {
      h[n] = h[n] * __expf(dt * A[n]) + dtx * sB[n];
      y += h[n] * sC[n];
    }
    yp[l] = y + Dv * x;
    __syncthreads();
  }
}

// Combine the 4 scan directions into y (B,DI,L)
__global__ void mmb_combine(const float* __restrict__ ys, float* __restrict__ y, int total) {
  int idx = blockIdx.x * blockDim.x + threadIdx.x;
  if (idx >= total) return;
  int l = idx % LL;
  int d = (idx / LL) % DI;
  int b = idx / (LL * DI);
  int hy = l / WW, wx = l % WW;
  int lt = wx * HH + hy;
  long base = (long)b * KK * DI * LL + (long)d * LL;
  float v = ys[base + 0L * DI * LL + l]
          + ys[base + 2L * DI * LL + (LL - 1 - l)]
          + ys[base + 1L * DI * LL + lt]
          + ys[base + 3L * DI * LL + (LL - 1 - lt)];
  y[((long)b * DI + d) * LL + l] = v;
}

// LayerNorm over channel dim (DI=192) at each (b,l), then multiply by z1.
// One wave per position; 6 channels per lane.
__global__ void mmb_ln_mul(const float* __restrict__ y, const float* __restrict__ g,
                           const float* __restrict__ bb, const float* __restrict__ z1,
                           float* __restrict__ out) {
  int gid = blockIdx.x * blockDim.x + threadIdx.x;
  int wavep = gid >> 5;
  int lane = gid & 31;
  if (wavep >= NB * LL) return;
  int b = wavep / LL, l = wavep % LL;
  const float* yp = y + (long)b * DI * LL + l;
  float v[6];
  float s = 0.f;
#pragma unroll
  for (int i = 0; i < 6; ++i) {
    v[i] = yp[(long)(i * 32 + lane) * LL];
    s += v[i];
  }
  for (int off = 16; off > 0; off >>= 1) s += __shfl_xor(s, off, 32);
  float m = s / (float)DI;
  float vs = 0.f;
#pragma unroll
  for (int i = 0; i < 6; ++i) {
    float c = v[i] - m;
    vs += c * c;
  }
  for (int off = 16; off > 0; off >>= 1) vs += __shfl_xor(vs, off, 32);
  float inv = rsqrtf(vs / (float)DI + 1e-5f);
#pragma unroll
  for (int i = 0; i < 6; ++i) {
    int d = i * 32 + lane;
    float o = (v[i] - m) * inv * g[d] + bb[d];
    out[((long)b * DI + d) * LL + l] = o * z1[((long)b * DI + d) * LL + l];
  }
}

// Per-(b,c) spatial mean: one wave each.
__global__ void mmb_mean(const float* __restrict__ x, float* __restrict__ gout) {
  int gid = blockIdx.x * blockDim.x + threadIdx.x;
  int w = gid >> 5;
  int lane = gid & 31;
  if (w >= NB * C2) return;
  const float* p = x + (long)w * LL;
  float s = 0.f;
  for (int i = lane; i < LL; i += 32) s += p[i];
  for (int off = 16; off > 0; off >>= 1) s += __shfl_xor(s, off, 32);
  if (lane == 0) gout[w] = s / (float)LL;
}

// Router: logits -> softmax(logits/2) -> top-2 normalized -> wmat (B x E)
__global__ void mmb_router(const float* __restrict__ gin, const float* __restrict__ rw,
                           const float* __restrict__ rb, float* __restrict__ wmat) {
  __shared__ float sl[NB * EE];
  int tid = threadIdx.x;
  if (tid < NB * EE) {
    int b = tid / EE, e = tid % EE;
    float s = rb[e];
    for (int c = 0; c < C2; ++c) s += gin[b * C2 + c] * rw[e * C2 + c];
    sl[tid] = s;
  }
  __syncthreads();
  if (tid < NB) {
    int b = tid;
    float lg[EE], p[EE];
    float mx = -1e30f;
    for (int e = 0; e < EE; ++e) {
      lg[e] = sl[b * EE + e] * 0.5f;
      mx = fmaxf(mx, lg[e]);
    }
    float ps = 0.f;
    for (int e = 0; e < EE; ++e) {
      p[e] = __expf(lg[e] - mx);
      ps += p[e];
    }
    for (int e = 0; e < EE; ++e) p[e] /= ps;
    int i0 = 0;
    for (int e = 1; e < EE; ++e) if (p[e] > p[i0]) i0 = e;
    int i1 = -1;
    for (int e = 0; e < EE; ++e) {
      if (e == i0) continue;
      if (i1 < 0 || p[e] > p[i1]) i1 = e;
    }
    float t = p[i0] + p[i1];
    for (int e = 0; e < EE; ++e) wmat[b * EE + e] = 0.f;
    wmat[b * EE + i0] = p[i0] / t;
    wmat[b * EE + i1] = p[i1] / t;
  }
}

__global__ void mmb_copy(const float* __restrict__ in, float* __restrict__ out, int total) {
  int idx = blockIdx.x * blockDim.x + threadIdx.x;
  if (idx >= total) return;
  out[idx] = in[idx];
}

// ---------------------------------------------------------------------------
extern "C" void kernel_launch(void* const* d_in, const int* in_sizes, int n_in,
                              void* d_out, int out_size, void* d_ws, size_t ws_size,
                              hipStream_t stream) {
  const float* x         = (const float*)d_in[0];
  const float* proj_w    = (const float*)d_in[1];
  const float* bn0_g     = (const float*)d_in[2];
  const float* bn0_b     = (const float*)d_in[3];
  const float* ls_fc1_w  = (const float*)d_in[4];
  const float* ls_fc1_b  = (const float*)d_in[5];
  const float* ls_bn_g   = (const float*)d_in[6];
  const float* ls_bn_b   = (const float*)d_in[7];
  const float* ls_fc2_w  = (const float*)d_in[8];
  const float* ls_fc2_b  = (const float*)d_in[9];
  const float* ls_fc3_w  = (const float*)d_in[10];
  const float* ls_fc3_b  = (const float*)d_in[11];
  const float* n1_g      = (const float*)d_in[12];
  const float* n1_b      = (const float*)d_in[13];
  const float* in_proj_w = (const float*)d_in[14];
  const float* conv_w    = (const float*)d_in[15];
  const float* conv_b    = (const float*)d_in[16];
  const float* x_proj_w  = (const float*)d_in[17];
  const float* dt_w      = (const float*)d_in[18];
  const float* dt_b      = (const float*)d_in[19];
  const float* A_logs    = (const float*)d_in[20];
  const float* Ds        = (const float*)d_in[21];
  const float* onorm_g   = (const float*)d_in[22];
  const float* onorm_b   = (const float*)d_in[23];
  const float* out_proj_w= (const float*)d_in[24];
  const float* n2_g      = (const float*)d_in[25];
  const float* n2_b      = (const float*)d_in[26];
  const float* router_w  = (const float*)d_in[27];
  const float* router_b  = (const float*)d_in[28];
  const float* moe_dw_w  = (const float*)d_in[29];
  const float* moe_dw_b  = (const float*)d_in[30];
  const float* moe_pw_w  = (const float*)d_in[31];
  const float* moe_pw_b  = (const float*)d_in[32];

  float* W = (float*)d_ws;
  const size_t SZ1   = (size_t)NB * C2 * LL;       // 614400
  const size_t SZxz  = (size_t)NB * 2 * DE * LL;   // 2457600
  const size_t SZd   = (size_t)NB * DE * LL;       // 1228800
  const size_t SZxs  = (size_t)NB * KK * DE * LL;  // 4915200
  const size_t SZxd  = (size_t)NB * KK * (RR + 2 * NN) * LL;  // 972800

  size_t o = 0;
  float* x_a  = W + o; o += SZ1;
  float* t1   = W + o; o += SZ1;
  float* t2   = W + o; o += SZ1;
  float* x1b  = W + o; o += SZ1;
  float* n1b  = W + o; o += SZ1;
  float* xz   = W + o; o += SZxz;
  float* xxb  = W + o; o += SZd;
  float* z1b  = W + o; o += SZd;
  float* xsb  = W + o; o += SZxs;
  float* xdbl = W + o; o += SZxd;
  float* dtsb = W + o; o += SZxs;
  float* Asb  = W + o; o += (size_t)KK * DI * NN;
  float* ysb  = W + o; o += SZxs;
  float* yb   = W + o; o += SZd;
  float* ylnb = W + o; o += SZd;
  float* xcur = W + o; o += SZ1;
  float* xm   = W + o; o += SZ1;
  float* he   = W + o; o += SZ1;
  float* gb   = W + o; o += (size_t)NB * C2;
  float* wm   = W + o; o += (size_t)NB * EE;

  auto cdiv = [](int a, int b) { return (a + b - 1) / b; };
  auto gemmGrid = [&](int M) { return cdiv(cdiv(M, 16) * ((NB * LL) / 32), 8); };

  const long sC2 = (long)C2 * LL;
  const long sDE = (long)DE * LL;
  const long sXZ = (long)2 * DE * LL;
  const long sXS = (long)KK * DE * LL;
  const long sXD = (long)KK * (RR + 2 * NN) * LL;

  // 1) x_a = silu(bn0(conv1x1(x, proj_w)))
  mmb_wmma_gemm<96, false><<<gemmGrid(C2), 256, 0, stream>>>(
      proj_w, x, x_a, C2, (long)C1 * LL, sC2, EPI_BNSILU, bn0_g, bn0_b,
      nullptr, 0, nullptr, 0);
  // 2) t1 = bn(dwconv3(x_a, ls_fc1) + b, ls_bn)
  mmb_dwconv3<<<cdiv(NB * C2 * LL, 256), 256, 0, stream>>>(
      x_a, t1, ls_fc1_w, ls_fc1_b, ls_bn_g, ls_bn_b, C2, sC2, sC2, 0, NB * C2 * LL);
  // 3) t2 = gelu(conv1x1(t1, ls_fc2) + b)
  mmb_wmma_gemm<96, false><<<gemmGrid(C2), 256, 0, stream>>>(
      ls_fc2_w, t1, t2, C2, sC2, sC2, EPI_BIAS_GELU, ls_fc2_b, nullptr,
      nullptr, 0, nullptr, 0);
  // 4) x1 = x_a + conv1x1(t2, ls_fc3) + b
  mmb_wmma_gemm<96, false><<<gemmGrid(C2), 256, 0, stream>>>(
      ls_fc3_w, t2, x1b, C2, sC2, sC2, EPI_BIAS_RES, ls_fc3_b, nullptr,
      x_a, sC2, nullptr, 0);
  // 5) n1 = bn(x1)
  mmb_bn<<<cdiv(NB * C2 * LL, 256), 256, 0, stream>>>(x1b, n1b, n1_g, n1_b, C2, NB * C2 * LL);
  // 6) xz = conv1x1(n1, in_proj_w)   (M = 384)
  mmb_wmma_gemm<96, false><<<gemmGrid(2 * DE), 256, 0, stream>>>(
      in_proj_w, n1b, xz, 2 * DE, sC2, sXZ, EPI_NONE, nullptr, nullptr,
      nullptr, 0, nullptr, 0);
  // 7) xx = gelu(dwconv3(xz[:, :DE], conv_w) + conv_b)
  mmb_dwconv3<<<cdiv(NB * DE * LL, 256), 256, 0, stream>>>(
      xz, xxb, conv_w, conv_b, nullptr, nullptr, DE, sXZ, sDE, 1, NB * DE * LL);
  // 8) z1 = gelu(xz[:, DE:])
  mmb_geluz<<<cdiv(NB * DE * LL, 256), 256, 0, stream>>>(xz, z1b, NB * DE * LL);
  // 9) xs (4 directions)
  mmb_build_xs<<<cdiv(NB * DE * LL, 256), 256, 0, stream>>>(xxb, xsb, NB * DE * LL);
  // 10) x_dbl[k] = x_proj_w[k] @ xs[:,k]   (M=38, K=192)
  for (int k = 0; k < KK; ++k)
    mmb_wmma_gemm<192, true><<<gemmGrid(RR + 2 * NN), 256, 0, stream>>>(
        x_proj_w + (size_t)k * (RR + 2 * NN) * DI, xsb + (size_t)k * DE * LL,
        xdbl + (size_t)k * (RR + 2 * NN) * LL, RR + 2 * NN, sXS, sXD,
        EPI_NONE, nullptr, nullptr, nullptr, 0, nullptr, 0);
  // 11) As = -exp(A_logs)
  mmb_make_as<<<cdiv(KK * DI * NN, 256), 256, 0, stream>>>(A_logs, Asb, KK * DI * NN);
  // 12) dts[k] = softplus(dt_w[k] @ x_dbl[:,k,:R] + dt_b[k])   (M=192, K=6)
  for (int k = 0; k < KK; ++k)
    mmb_wmma_gemm<6, false><<<gemmGrid(DI), 256, 0, stream>>>(
        dt_w + (size_t)k * DI * RR, xdbl + (size_t)k * (RR + 2 * NN) * LL,
        dtsb + (size_t)k * DI * LL, DI, sXD, sXS,
        EPI_BIAS_SOFTPLUS, dt_b + (size_t)k * DI, nullptr, nullptr, 0, nullptr, 0);
  // 13) selective scan
  mmb_scan<<<NB * KK, DI, 0, stream>>>(xsb, dtsb, xdbl, Asb, Ds, ysb);
  // 14) combine directions
  mmb_combine<<<cdiv(NB * DI * LL, 256), 256, 0, stream>>>(ysb, yb, NB * DI * LL);
  // 15) layernorm over D, * z1
  mmb_ln_mul<<<cdiv(NB * LL * 32, 256), 256, 0, stream>>>(yb, onorm_g, onorm_b, z1b, ylnb);
  // 16) xcur = x_a + conv1x1(yln, out_proj_w)   (K=192)
  mmb_wmma_gemm<192, false><<<gemmGrid(C2), 256, 0, stream>>>(
      out_proj_w, ylnb, xcur, C2, sDE, sC2, EPI_RES, nullptr, nullptr,
      x_a, sC2, nullptr, 0);
  // 17) xm = bn(xcur, n2)
  mmb_bn<<<cdiv(NB * C2 * LL, 256), 256, 0, stream>>>(xcur, xm, n2_g, n2_b, C2, NB * C2 * LL);
  // 18) channel means, 19) router
  mmb_mean<<<cdiv(NB * C2 * 32, 256), 256, 0, stream>>>(xm, gb);
  mmb_router<<<1, 32, 0, stream>>>(gb, router_w, router_b, wm);
  // 20) out = xcur
  float* outp = (float*)d_out;
  mmb_copy<<<cdiv(NB * C2 * LL, 256), 256, 0, stream>>>(xcur, outp, NB * C2 * LL);
  // 21) MoE experts: out += wmat[b,e] * (conv1x1(gelu(dwconv3(xm)), pw) + b)
  for (int e = 0; e < EE; ++e) {
    mmb_dwconv3<<<cdiv(NB * C2 * LL, 256), 256, 0, stream>>>(
        xm, he, moe_dw_w + (size_t)e * C2 * 9, moe_dw_b + (size_t)e * C2,
        nullptr, nullptr, C2, sC2, sC2, 1, NB * C2 * LL);
    mmb_wmma_gemm<96, false><<<gemmGrid(C2), 256, 0, stream>>>(
        moe_pw_w + (size_t)e * C2 * C2, he, outp, C2, sC2, sC2,
        EPI_MOE, moe_pw_b + (size_t)e * C2, nullptr, nullptr, 0, wm, e);
  }
  (void)in_sizes; (void)n_in; (void)out_size; (void)ws_size;
}